// KWTAMask_11940009083437
// MI455X (gfx1250) — compile-verified
//
#include <hip/hip_runtime.h>
#include <stdint.h>

typedef __attribute__((ext_vector_type(4))) float v4f;
typedef __attribute__((ext_vector_type(4))) int   v4i;

#define KSEL      100000u
#define NBINS     4096
#define IDX_PFX   4096
#define IDX_KREM  4097
#define IDX_THR   4098

#if defined(__HIP_DEVICE_COMPILE__) && defined(__gfx1250__)
  #if __has_builtin(__builtin_amdgcn_global_load_async_to_lds_b128)
    #define USE_ASYNC_LDS 1
    typedef __attribute__((address_space(1))) v4i* gptr_v4i;
    typedef __attribute__((address_space(3))) v4i* lptr_v4i;
  #endif
  #if __has_builtin(__builtin_amdgcn_s_wait_asynccnt)
    #define WAIT_ASYNC(n) __builtin_amdgcn_s_wait_asynccnt(n)
  #else
    #define WAIT_ASYNC(n) asm volatile("s_wait_asynccnt %0" :: "i"(n) : "memory")
  #endif
#else
  #define WAIT_ASYNC(n)
#endif

// Monotonic bijection f32 -> u32: float order == unsigned order.
__device__ __forceinline__ uint32_t fkey(float f) {
  uint32_t u = __float_as_uint(f);
  return (u & 0x80000000u) ? ~u : (u | 0x80000000u);
}

// Histogram pass (defined FIRST so the disasm snippet shows the async path).
// level 0: bins = key[31:20]; level 1: key[19:8] where key[31:20]==pfx;
// level 2: key[7:0] where key[31:8]==pfx.
__global__ __launch_bounds__(256) void kwta_hist(const float* __restrict__ x,
                                                 long long n,
                                                 uint32_t* __restrict__ W,
                                                 int level) {
  __shared__ uint32_t lhist[2][NBINS];          // 2 privatized copies: 32 KB
#if USE_ASYNC_LDS
  __shared__ float stage[2][1024];              // double-buffered tile: 8 KB
#endif
  const int tid = threadIdx.x;
  for (int i = tid; i < NBINS; i += 256) { lhist[0][i] = 0u; lhist[1][i] = 0u; }
  __syncthreads();

  const uint32_t pfx = (level == 0) ? 0u : W[IDX_PFX];
  uint32_t* myh = lhist[tid & 1];

  auto tally = [&](float f) {
    uint32_t key = fkey(f);
    if (level == 0) {
      atomicAdd(&myh[key >> 20], 1u);
    } else if (level == 1) {
      if ((key >> 20) == pfx) atomicAdd(&myh[(key >> 8) & 0xFFFu], 1u);
    } else {
      if ((key >> 8) == pfx) atomicAdd(&myh[key & 0xFFu], 1u);
    }
  };

  const long long numTiles = n >> 10;           // 1024 floats per block-tile
  const long long gstride  = gridDim.x;
  long long tile = blockIdx.x;

#if USE_ASYNC_LDS
  // Each lane async-copies its own 16B to its own LDS slot; per-wave
  // s_wait_asynccnt is the only sync needed (no cross-lane consumption).
  int buf = 0;
  if (tile < numTiles) {
    __builtin_amdgcn_global_load_async_to_lds_b128(
        (gptr_v4i)(x + (tile << 10) + tid * 4),
        (lptr_v4i)(&stage[0][tid * 4]), 0, 0);
  }
  for (; tile < numTiles; tile += gstride) {
    const long long nxt = tile + gstride;
    if (nxt < numTiles) {
      __builtin_amdgcn_global_load_async_to_lds_b128(
          (gptr_v4i)(x + (nxt << 10) + tid * 4),
          (lptr_v4i)(&stage[buf ^ 1][tid * 4]), 0, 0);
      WAIT_ASYNC(1);                            // oldest stage done (in-order)
    } else {
      WAIT_ASYNC(0);
    }
    v4f v = *(const v4f*)&stage[buf][tid * 4];
    tally(v.x); tally(v.y); tally(v.z); tally(v.w);
    buf ^= 1;
  }
#else
  for (; tile < numTiles; tile += gstride) {
    v4f v = *(const v4f*)(x + (tile << 10) + tid * 4);
    tally(v.x); tally(v.y); tally(v.z); tally(v.w);
  }
#endif

  // Tail (n not multiple of 1024)
  for (long long i = (numTiles << 10) + (long long)blockIdx.x * 256 + tid;
       i < n; i += gstride * 256)
    tally(x[i]);

  __syncthreads();
  for (int i = tid; i < NBINS; i += 256) {
    uint32_t c = lhist[0][i] + lhist[1][i];
    if (c) atomicAdd(&W[i], c);
  }
}

__global__ void kwta_zero(uint32_t* __restrict__ W) {
  int i = blockIdx.x * blockDim.x + threadIdx.x;
  if (i < NBINS) W[i] = 0u;
}

// Single-block selection: find the bin (descending key order) containing the
// K-th largest element; update prefix / remaining-K / final threshold key.
__global__ __launch_bounds__(256) void kwta_select(uint32_t* __restrict__ W,
                                                   int nbins, int level) {
  __shared__ uint32_t partial[256];
  __shared__ uint32_t sChunk, sBefore, sK;
  const int t = threadIdx.x;
  const int per = nbins >> 8;                   // 16 (4096 bins) or 1 (256)
  uint32_t s = 0;
  for (int i = 0; i < per; ++i) s += W[t * per + i];
  partial[t] = s;
  __syncthreads();

  if (t == 0) {
    uint32_t K = (level == 0) ? KSEL : W[IDX_KREM];
    uint32_t cum = 0; int cb = 0;
    for (int b = 255; b >= 0; --b) {
      if (cum + partial[b] >= K) { cb = b; break; }
      cum += partial[b];
    }
    sChunk = (uint32_t)cb; sBefore = cum; sK = K;
  }
  __syncthreads();

  if (t == (int)sChunk) {
    uint32_t K = sK, cum = sBefore, kr = 1;
    int bin = t * per;
    for (int i = per - 1; i >= 0; --i) {
      uint32_t c = W[t * per + i];
      if (cum + c >= K) { bin = t * per + i; kr = K - cum; break; }
      cum += c;
    }
    if (level == 0)      { W[IDX_PFX] = (uint32_t)bin;                 W[IDX_KREM] = kr; }
    else if (level == 1) { W[IDX_PFX] = (W[IDX_PFX] << 12) | (uint32_t)bin; W[IDX_KREM] = kr; }
    else                 { W[IDX_THR] = (W[IDX_PFX] << 8)  | (uint32_t)bin; }
  }
}

// Final streamed mask: out[i] = (key(x[i]) >= thrKey). Non-temporal (last use;
// don't evict useful L2 lines with the 134MB output stream).
__global__ __launch_bounds__(256) void kwta_mask(const float* __restrict__ x,
                                                 float* __restrict__ out,
                                                 long long n,
                                                 const uint32_t* __restrict__ W) {
  const uint32_t thr = W[IDX_THR];
  const long long n4 = n >> 2;
  const long long stride = (long long)gridDim.x * blockDim.x;
  for (long long i = (long long)blockIdx.x * blockDim.x + threadIdx.x;
       i < n4; i += stride) {
    v4f v = __builtin_nontemporal_load((const v4f*)x + i);
    v4f r;
    r.x = (fkey(v.x) >= thr) ? 1.0f : 0.0f;
    r.y = (fkey(v.y) >= thr) ? 1.0f : 0.0f;
    r.z = (fkey(v.z) >= thr) ? 1.0f : 0.0f;
    r.w = (fkey(v.w) >= thr) ? 1.0f : 0.0f;
    __builtin_nontemporal_store(r, (v4f*)out + i);
  }
  for (long long i = (n4 << 2) + (long long)blockIdx.x * blockDim.x + threadIdx.x;
       i < n; i += stride)
    out[i] = (fkey(x[i]) >= thr) ? 1.0f : 0.0f;
}

extern "C" void kernel_launch(void* const* d_in, const int* in_sizes, int n_in,
                              void* d_out, int out_size, void* d_ws, size_t ws_size,
                              hipStream_t stream) {
  const float* x = (const float*)d_in[0];
  float* out = (float*)d_out;
  uint32_t* W = (uint32_t*)d_ws;                // 4096 bins + 3 state words
  const long long n = (long long)in_sizes[0];

  const dim3 blk(256);
  const dim3 histGrid(4096);                    // grid-stride, ~8 tiles/block
  const dim3 maskGrid(8192);

  for (int level = 0; level < 3; ++level) {
    hipLaunchKernelGGL(kwta_zero,   dim3(16),  blk, 0, stream, W);
    hipLaunchKernelGGL(kwta_hist,   histGrid,  blk, 0, stream, x, n, W, level);
    hipLaunchKernelGGL(kwta_select, dim3(1),   blk, 0, stream, W,
                       (level == 2) ? 256 : NBINS, level);
  }
  hipLaunchKernelGGL(kwta_mask, maskGrid, blk, 0, stream, x, out, n, W);
}